// Net_22196390986357
// MI455X (gfx1250) — compile-verified
//
#include <hip/hip_runtime.h>

// ---------------------------------------------------------------------------
// 2-layer GATConv for MI455X (gfx1250, wave32, WMMA).
//   L1: h1 = x(100000x256) @ W1(256x64)  -> WMMA f32 16x16x4 pipeline with
//       async global->LDS double-buffered staging (ASYNCcnt path).
//   edge passes (3.3M edges incl. self-loops, 8 heads): seg-max / seg-sum /
//   weighted aggregate via L2-resident atomics.
//   L2: tiny 64x3 GEMM + single-head edge passes.
// ---------------------------------------------------------------------------

#define NEG_SLOPE 0.2f
#define EPSV 1e-16f

typedef __attribute__((ext_vector_type(2))) float v2f;
typedef __attribute__((ext_vector_type(8))) float v8f;

// Order-preserving encoding of f32 into u32 so unsigned atomicMax == float max.
__device__ __forceinline__ unsigned fenc(float f) {
  unsigned u = __float_as_uint(f);
  return (u & 0x80000000u) ? ~u : (u | 0x80000000u);
}
__device__ __forceinline__ float fdec(unsigned u) {
  return __uint_as_float((u & 0x80000000u) ? (u & 0x7fffffffu) : ~u);
}

// CDNA5 async global->LDS copy (ASYNCcnt), 16 bytes per lane.
// lds_off = low 32 bits of the flat address of the LDS destination.
__device__ __forceinline__ void async_ld_b128(unsigned lds_off, const void* gptr) {
  asm volatile("global_load_async_to_lds_b128 %0, %1, off"
               :: "v"(lds_off), "v"((unsigned long long)(uintptr_t)gptr)
               : "memory");
}
__device__ __forceinline__ void wait_async0() {
  asm volatile("s_wait_asynccnt 0" ::: "memory");
}

// ---------------------------------------------------------------------------
// GEMM1: h1[N,64] = x[N,256] @ W1[256,64], f32 WMMA 16x16x4.
// Block = 256 threads = 8 waves; each block does a 32x64 output tile,
// waves arranged 2 (M) x 4 (N). K staged through LDS in chunks of 64 using
// double-buffered GLOBAL_LOAD_ASYNC_TO_LDS_B128.
// ---------------------------------------------------------------------------
__global__ __launch_bounds__(256) void gat_gemm1(const float* __restrict__ x,
                                                 const float* __restrict__ W,
                                                 float* __restrict__ h,
                                                 int n_nodes) {
  __shared__ float As[2][32 * 64];   // 2 x 8 KB
  __shared__ float Bs[2][64 * 64];   // 2 x 16 KB
  const int tid = threadIdx.x;
  const int m0 = blockIdx.x * 32;
  const int wave = tid >> 5;      // wave32
  const int lane = tid & 31;
  const int laneHi = lane >> 4;   // selects K pair {0,1} vs {2,3}
  const int lane16 = lane & 15;
  const int mt = wave >> 2;       // 0..1
  const int nt = wave & 3;        // 0..3
  const int arow = mt * 16 + lane16;   // A: both lane halves hold M=0..15
  const int bcol = nt * 16 + lane16;   // B: N = lane%16

  const unsigned as_base = (unsigned)(uintptr_t)&As[0][0];
  const unsigned bs_base = (unsigned)(uintptr_t)&Bs[0][0];
  const unsigned as_buf_bytes = 32 * 64 * 4;
  const unsigned bs_buf_bytes = 64 * 64 * 4;

  // Issue async staging of one 64-wide K chunk into buffer `buf`.
  auto issue = [&](int kc, int buf) {
    // A chunk: 32 rows x 64 cols = 512 float4 slots, 2 per thread.
    for (int t = 0; t < 2; ++t) {
      int i = tid + t * 256;
      int row = i >> 4, c4 = i & 15;
      int gr = m0 + row;
      if (gr >= n_nodes) gr = n_nodes - 1;   // clamp: keep EXEC full for WMMA
      async_ld_b128(as_base + buf * as_buf_bytes + i * 16,
                    x + 4 * ((size_t)gr * 64 + (kc >> 2) + c4));
    }
    // B chunk: 64 rows x 64 cols = 1024 float4 slots, 4 per thread.
    for (int t = 0; t < 4; ++t) {
      int j = tid + t * 256;
      int row = j >> 4, c4 = j & 15;
      async_ld_b128(bs_base + buf * bs_buf_bytes + j * 16,
                    W + 4 * ((size_t)(kc + row) * 16 + c4));
    }
  };

  issue(0, 0);
  v8f acc = {};
  for (int kc = 0; kc < 256; kc += 64) {
    const int buf = (kc >> 6) & 1;
    wait_async0();        // my async LDS writes (incl. this chunk) done
    __syncthreads();      // everyone's done -> chunk visible; prev reads done
    if (kc + 64 < 256) issue(kc + 64, buf ^ 1);
    const float* A = &As[buf][0];
    const float* B = &Bs[buf][0];
    for (int k = 0; k < 64; k += 4) {
      const int ka = k + 2 * laneHi;
      v2f a = *(const v2f*)&A[arow * 64 + ka];      // K = ka, ka+1
      v2f b;
      b.x = B[(ka + 0) * 64 + bcol];
      b.y = B[(ka + 1) * 64 + bcol];
      acc = __builtin_amdgcn_wmma_f32_16x16x4_f32(false, a, false, b,
                                                  (short)0, acc, false, false);
    }
  }
  // C/D layout: VGPR r -> M = r + 8*laneHi, N = lane%16
  const int grow = m0 + mt * 16 + laneHi * 8;
  const int gcol = nt * 16 + lane16;
  for (int r = 0; r < 8; ++r) {
    int gr = grow + r;
    if (gr < n_nodes) h[(size_t)gr * 64 + gcol] = acc[r];
  }
}

// ---------------------------------------------------------------------------
// Per-node attention coefficients, layer 1: alpha[n,h] = sum_c h[n,h,c]*a[h,c]
// ---------------------------------------------------------------------------
__global__ void gat_alpha1(const float* __restrict__ h,
                           const float* __restrict__ a_src,
                           const float* __restrict__ a_dst,
                           float* __restrict__ as, float* __restrict__ ad,
                           int n_nodes) {
  int i = blockIdx.x * blockDim.x + threadIdx.x;
  if (i >= n_nodes * 8) return;
  const int hh = i & 7;
  const float* hp = h + (size_t)(i >> 3) * 64 + hh * 8;
  float ss = 0.f, dd = 0.f;
  for (int c = 0; c < 8; ++c) {
    float v = hp[c];
    ss += v * a_src[hh * 8 + c];
    dd += v * a_dst[hh * 8 + c];
  }
  as[i] = ss;
  ad[i] = dd;
}

__device__ __forceinline__ void edge_sd(const int* __restrict__ src,
                                        const int* __restrict__ dst,
                                        long long e, long long E,
                                        int& s, int& d) {
  if (e < E) { s = src[e]; d = dst[e]; }
  else       { s = d = (int)(e - E); }   // implicit self-loop
}

// ------------------------ layer-1 edge passes (8 heads) --------------------
__global__ void gat_edge_max1(const int* __restrict__ src, const int* __restrict__ dst,
                              long long E, long long tot,
                              const float* __restrict__ as, const float* __restrict__ ad,
                              unsigned* __restrict__ menc) {
  long long i = (long long)blockIdx.x * blockDim.x + threadIdx.x;
  if (i >= tot * 8) return;
  long long e = i >> 3; int hh = (int)(i & 7);
  int s, d; edge_sd(src, dst, e, E, s, d);
  float ev = as[(size_t)s * 8 + hh] + ad[(size_t)d * 8 + hh];
  ev = ev > 0.f ? ev : NEG_SLOPE * ev;
  atomicMax(&menc[(size_t)d * 8 + hh], fenc(ev));
}

__global__ void gat_edge_sum1(const int* __restrict__ src, const int* __restrict__ dst,
                              long long E, long long tot,
                              const float* __restrict__ as, const float* __restrict__ ad,
                              const unsigned* __restrict__ menc,
                              float* __restrict__ ssum) {
  long long i = (long long)blockIdx.x * blockDim.x + threadIdx.x;
  if (i >= tot * 8) return;
  long long e = i >> 3; int hh = (int)(i & 7);
  int s, d; edge_sd(src, dst, e, E, s, d);
  float ev = as[(size_t)s * 8 + hh] + ad[(size_t)d * 8 + hh];
  ev = ev > 0.f ? ev : NEG_SLOPE * ev;
  float ex = expf(ev - fdec(menc[(size_t)d * 8 + hh]));
  atomicAdd(&ssum[(size_t)d * 8 + hh], ex);
}

__global__ void gat_edge_aggr1(const int* __restrict__ src, const int* __restrict__ dst,
                               long long E, long long tot,
                               const float* __restrict__ as, const float* __restrict__ ad,
                               const unsigned* __restrict__ menc,
                               const float* __restrict__ ssum,
                               const float* __restrict__ h, float* __restrict__ out) {
  long long i = (long long)blockIdx.x * blockDim.x + threadIdx.x;
  if (i >= tot * 8) return;
  long long e = i >> 3; int hh = (int)(i & 7);
  int s, d; edge_sd(src, dst, e, E, s, d);
  float ev = as[(size_t)s * 8 + hh] + ad[(size_t)d * 8 + hh];
  ev = ev > 0.f ? ev : NEG_SLOPE * ev;
  float ex = expf(ev - fdec(menc[(size_t)d * 8 + hh]));
  float a = ex / (ssum[(size_t)d * 8 + hh] + EPSV);
  const float* hp = h + (size_t)s * 64 + hh * 8;
  float* op = out + (size_t)d * 64 + hh * 8;
  for (int c = 0; c < 8; ++c) atomicAdd(&op[c], hp[c] * a);
}

// bias + ELU in place
__global__ void gat_finish1(float* __restrict__ out1, const float* __restrict__ b,
                            long long n64) {
  long long i = (long long)blockIdx.x * blockDim.x + threadIdx.x;
  if (i >= n64) return;
  float v = out1[i] + b[i & 63];
  out1[i] = v > 0.f ? v : expm1f(v);
}

// ---------------------------------------------------------------------------
// GEMM2 (64x3, trivial) fused with layer-2 alpha computation.
// ---------------------------------------------------------------------------
__global__ void gat_gemm2(const float* __restrict__ act, const float* __restrict__ W2,
                          const float* __restrict__ a_src2, const float* __restrict__ a_dst2,
                          float* __restrict__ h2, float* __restrict__ as2,
                          float* __restrict__ ad2, int n_nodes) {
  int n = blockIdx.x * blockDim.x + threadIdx.x;
  if (n >= n_nodes) return;
  const float* ap = act + (size_t)n * 64;
  float s0 = 0.f, s1 = 0.f, s2 = 0.f;
  for (int k = 0; k < 64; ++k) {
    float v = ap[k];
    s0 += v * W2[k * 3 + 0];
    s1 += v * W2[k * 3 + 1];
    s2 += v * W2[k * 3 + 2];
  }
  h2[(size_t)n * 3 + 0] = s0;
  h2[(size_t)n * 3 + 1] = s1;
  h2[(size_t)n * 3 + 2] = s2;
  as2[n] = s0 * a_src2[0] + s1 * a_src2[1] + s2 * a_src2[2];
  ad2[n] = s0 * a_dst2[0] + s1 * a_dst2[1] + s2 * a_dst2[2];
}

// ------------------------ layer-2 edge passes (1 head) ---------------------
__global__ void gat_edge_max2(const int* __restrict__ src, const int* __restrict__ dst,
                              long long E, long long tot,
                              const float* __restrict__ as, const float* __restrict__ ad,
                              unsigned* __restrict__ menc) {
  long long e = (long long)blockIdx.x * blockDim.x + threadIdx.x;
  if (e >= tot) return;
  int s, d; edge_sd(src, dst, e, E, s, d);
  float ev = as[s] + ad[d];
  ev = ev > 0.f ? ev : NEG_SLOPE * ev;
  atomicMax(&menc[d], fenc(ev));
}

__global__ void gat_edge_sum2(const int* __restrict__ src, const int* __restrict__ dst,
                              long long E, long long tot,
                              const float* __restrict__ as, const float* __restrict__ ad,
                              const unsigned* __restrict__ menc, float* __restrict__ ssum) {
  long long e = (long long)blockIdx.x * blockDim.x + threadIdx.x;
  if (e >= tot) return;
  int s, d; edge_sd(src, dst, e, E, s, d);
  float ev = as[s] + ad[d];
  ev = ev > 0.f ? ev : NEG_SLOPE * ev;
  atomicAdd(&ssum[d], expf(ev - fdec(menc[d])));
}

__global__ void gat_edge_aggr2(const int* __restrict__ src, const int* __restrict__ dst,
                               long long E, long long tot,
                               const float* __restrict__ as, const float* __restrict__ ad,
                               const unsigned* __restrict__ menc,
                               const float* __restrict__ ssum,
                               const float* __restrict__ h2, float* __restrict__ out) {
  long long e = (long long)blockIdx.x * blockDim.x + threadIdx.x;
  if (e >= tot) return;
  int s, d; edge_sd(src, dst, e, E, s, d);
  float ev = as[s] + ad[d];
  ev = ev > 0.f ? ev : NEG_SLOPE * ev;
  float ex = expf(ev - fdec(menc[d]));
  float a = ex / (ssum[d] + EPSV);
  for (int c = 0; c < 3; ++c)
    atomicAdd(&out[(size_t)d * 3 + c], h2[(size_t)s * 3 + c] * a);
}

__global__ void gat_finish2(float* __restrict__ out, const float* __restrict__ b2,
                            int n_nodes) {
  int i = blockIdx.x * blockDim.x + threadIdx.x;
  if (i >= n_nodes * 3) return;
  out[i] += b2[i % 3];
}

// ---------------------------------------------------------------------------
extern "C" void kernel_launch(void* const* d_in, const int* in_sizes, int n_in,
                              void* d_out, int out_size, void* d_ws, size_t ws_size,
                              hipStream_t stream) {
  const float* x      = (const float*)d_in[0];
  const int*   ei     = (const int*)d_in[1];
  const float* W1     = (const float*)d_in[2];
  const float* a_src1 = (const float*)d_in[3];
  const float* a_dst1 = (const float*)d_in[4];
  const float* b1     = (const float*)d_in[5];
  const float* W2     = (const float*)d_in[6];
  const float* a_src2 = (const float*)d_in[7];
  const float* a_dst2 = (const float*)d_in[8];
  const float* b2     = (const float*)d_in[9];

  const int N = in_sizes[0] / 256;
  const long long E = in_sizes[1] / 2;
  const int* src = ei;
  const int* dst = ei + E;
  const long long tot = E + (long long)N;   // with implicit self-loops

  // workspace carve-up (floats)
  float*    h1   = (float*)d_ws;                       // N*64
  float*    as1  = h1  + (size_t)N * 64;               // N*8
  float*    ad1  = as1 + (size_t)N * 8;                // N*8
  unsigned* m1   = (unsigned*)(ad1 + (size_t)N * 8);   // N*8
  float*    s1   = (float*)(m1 + (size_t)N * 8);       // N*8
  float*    out1 = s1  + (size_t)N * 8;                // N*64 (becomes act1)
  float*    h2   = out1 + (size_t)N * 64;              // N*3
  float*    as2  = h2  + (size_t)N * 3;                // N
  float*    ad2  = as2 + (size_t)N;                    // N
  unsigned* m2   = (unsigned*)(ad2 + (size_t)N);       // N
  float*    s2   = (float*)(m2 + (size_t)N);           // N
  float*    outv = (float*)d_out;                      // N*3

  // zero accumulators (fenc: any finite value encodes > 0, so 0 == -inf)
  (void)hipMemsetAsync(m1,   0, (size_t)N * 8 * sizeof(unsigned), stream);
  (void)hipMemsetAsync(s1,   0, (size_t)N * 8 * sizeof(float), stream);
  (void)hipMemsetAsync(out1, 0, (size_t)N * 64 * sizeof(float), stream);
  (void)hipMemsetAsync(m2,   0, (size_t)N * sizeof(unsigned), stream);
  (void)hipMemsetAsync(s2,   0, (size_t)N * sizeof(float), stream);
  (void)hipMemsetAsync(d_out, 0, (size_t)out_size * sizeof(float), stream);

  gat_gemm1<<<(N + 31) / 32, 256, 0, stream>>>(x, W1, h1, N);
  gat_alpha1<<<(unsigned)(((size_t)N * 8 + 255) / 256), 256, 0, stream>>>(
      h1, a_src1, a_dst1, as1, ad1, N);

  const unsigned blocks1 = (unsigned)((tot * 8 + 255) / 256);
  gat_edge_max1 <<<blocks1, 256, 0, stream>>>(src, dst, E, tot, as1, ad1, m1);
  gat_edge_sum1 <<<blocks1, 256, 0, stream>>>(src, dst, E, tot, as1, ad1, m1, s1);
  gat_edge_aggr1<<<blocks1, 256, 0, stream>>>(src, dst, E, tot, as1, ad1, m1, s1, h1, out1);
  gat_finish1<<<(unsigned)(((size_t)N * 64 + 255) / 256), 256, 0, stream>>>(
      out1, b1, (long long)N * 64);

  gat_gemm2<<<(N + 255) / 256, 256, 0, stream>>>(out1, W2, a_src2, a_dst2, h2, as2, ad2, N);

  const unsigned blocks2 = (unsigned)((tot + 255) / 256);
  gat_edge_max2 <<<blocks2, 256, 0, stream>>>(src, dst, E, tot, as2, ad2, m2);
  gat_edge_sum2 <<<blocks2, 256, 0, stream>>>(src, dst, E, tot, as2, ad2, m2, s2);
  gat_edge_aggr2<<<blocks2, 256, 0, stream>>>(src, dst, E, tot, as2, ad2, m2, s2, h2, outv);
  gat_finish2<<<(unsigned)(((size_t)N * 3 + 255) / 256), 256, 0, stream>>>(outv, b2, N);
}